// SkipGramModel_40364102647844
// MI455X (gfx1250) — compile-verified
//
#include <hip/hip_runtime.h>
#include <hip/hip_bf16.h>

typedef __attribute__((ext_vector_type(2))) float v2f;
typedef __attribute__((ext_vector_type(8))) float v8f;

#define C_LEN   200
#define D_DIM   128
#define R_TOTAL (C_LEN + C_LEN * D_DIM)          // 25800 gathered rows total
#define GROUPS  ((R_TOTAL + 15) / 16)            // 1613 groups of 16 rows
#define WAVES_PER_BLOCK 8
#define BLOCK   (WAVES_PER_BLOCK * 32)           // wave32

__global__ void sg_zero_out(float* out) {
    if (threadIdx.x == 0) out[0] = 0.0f;
}

__device__ __forceinline__ float log_sigmoid(float x) {
    // numerically stable: min(x,0) - log1p(exp(-|x|))
    return fminf(x, 0.0f) - log1pf(__expf(-fabsf(x)));
}

__global__ __launch_bounds__(BLOCK) void sg_loss_kernel(
    const float* __restrict__ W_in,
    const float* __restrict__ W_out,
    const int*   __restrict__ target,
    const int*   __restrict__ context,
    const int*   __restrict__ negs,
    float*       __restrict__ out)
{
    __shared__ float t_sh[D_DIM];

    const int tid = threadIdx.x;
    if (tid < D_DIM) {
        const float* t = W_in + (size_t)target[0] * D_DIM;
        t_sh[tid] = t[tid];
    }
    __syncthreads();

    const int wave  = tid >> 5;
    const int lane  = tid & 31;
    const int group = blockIdx.x * WAVES_PER_BLOCK + wave;
    if (group >= GROUPS) return;                 // uniform per wave: EXEC stays all-ones

    const int base = group * 16;

    // A-fragment row this lane feeds (ISA 16x4 f32 A layout):
    //   lanes 0-15: M=lane,   holds K=0,1  -> koff 0
    //   lanes16-31: M=lane-16,holds K=2,3  -> koff 2
    const int koff = (lane >> 4) << 1;
    int rowA = base + (lane & 15);
    rowA = (rowA < R_TOTAL) ? rowA : 0;          // clamp; tail masked later per-accumulator
    const int idx = (rowA < C_LEN) ? context[rowA] : negs[rowA - C_LEN];
    const float* rowp = W_out + (size_t)idx * D_DIM + koff;

    v8f acc0 = {};
    v8f acc1 = {};

#pragma unroll
    for (int s = 0; s < 32; s += 2) {
        // A: 2 consecutive f32 of this lane's row at K = 4*s + koff
        v2f a0 = *(const v2f*)(rowp + 4 * s);
        // B: t broadcast over all N columns; lane supplies B[k][n] with k = koff + {0,1}
        v2f b0;
        b0.x = t_sh[4 * s + koff];
        b0.y = t_sh[4 * s + koff + 1];
        acc0 = __builtin_amdgcn_wmma_f32_16x16x4_f32(
            false, a0, false, b0, (short)0, acc0, false, false);

        v2f a1 = *(const v2f*)(rowp + 4 * (s + 1));
        v2f b1;
        b1.x = t_sh[4 * (s + 1) + koff];
        b1.y = t_sh[4 * (s + 1) + koff + 1];
        acc1 = __builtin_amdgcn_wmma_f32_16x16x4_f32(
            false, a1, false, b1, (short)0, acc1, false, false);
    }

    // D layout (32-bit C/D 16x16): VGPR j, lanes 0-15 -> M=j; lanes 16-31 -> M=j+8.
    // Each row's score is replicated across the 16 N columns -> divide by 16 at the end.
    const int mhalf = (lane >> 4) << 3;
    float local = 0.0f;
#pragma unroll
    for (int j = 0; j < 8; ++j) {
        const int row = base + j + mhalf;
        const float s = acc0[j] + acc1[j];
        const float x = (row < C_LEN) ? s : -s;  // positives: +score, negatives: -score
        const float ls = log_sigmoid(x);
        local += (row < R_TOTAL) ? ls : 0.0f;    // mask tail rows
    }

    // wave32 reduction
#pragma unroll
    for (int off = 16; off > 0; off >>= 1)
        local += __shfl_xor(local, off, 32);

    if (lane == 0)
        atomicAdd(out, -local * (1.0f / 16.0f)); // 16x replication across N; loss = -(sum)
}

extern "C" void kernel_launch(void* const* d_in, const int* in_sizes, int n_in,
                              void* d_out, int out_size, void* d_ws, size_t ws_size,
                              hipStream_t stream) {
    const float* W_in    = (const float*)d_in[0];
    const float* W_out   = (const float*)d_in[1];
    const int*   target  = (const int*)d_in[2];
    const int*   context = (const int*)d_in[3];
    const int*   negs    = (const int*)d_in[4];
    float*       out     = (float*)d_out;

    sg_zero_out<<<1, 64, 0, stream>>>(out);

    const int blocks = (GROUPS + WAVES_PER_BLOCK - 1) / WAVES_PER_BLOCK;
    sg_loss_kernel<<<blocks, BLOCK, 0, stream>>>(W_in, W_out, target, context, negs, out);
}